// Attention_88467736363354
// MI455X (gfx1250) — compile-verified
//
#include <hip/hip_runtime.h>
#include <hip/hip_bf16.h>

typedef float v2f __attribute__((ext_vector_type(2)));
typedef float v8f __attribute__((ext_vector_type(8)));

#define BATCH 256
#define SEQ   1024
#define CIN   256
#define DHEAD 64
#define NH    8

// ---------------------------------------------------------------------------
// Kernel 1: per (head, 16-batch tile): q = queries@Wq^T + bq  (B,H,64)
//           qk[b,h,i] = sum_d q[b,h,d]*Wk[h,d,i]   (B,H,256)
//           qb[b,h]   = q[b,h,:]·bk[h,:]
// Tiny (~100 MFLOP total); Wq/Wk are L2-resident (512 KB).
// ---------------------------------------------------------------------------
__global__ __launch_bounds__(256) void qk_prep_kernel(
    const float* __restrict__ queries,
    const float* __restrict__ Wq, const float* __restrict__ bq,
    const float* __restrict__ Wk, const float* __restrict__ bk,
    float* __restrict__ qk_ws, float* __restrict__ qb_ws)
{
    const int h  = blockIdx.x & 7;
    const int b0 = (blockIdx.x >> 3) * 16;
    const int t  = threadIdx.x;

    __shared__ float qtile[16][CIN];   // 16 KB: queries slice
    __shared__ float q16[16][DHEAD];   // 4 KB: projected q

    // stage queries[b0..b0+16) (coalesced float4)
    for (int idx = t * 4; idx < 16 * CIN; idx += 256 * 4)
        *(float4*)(&qtile[0][0] + idx) = *(const float4*)(queries + b0 * CIN + idx);
    __syncthreads();

    // q16[bl][d] = bq + queries[b]·Wq[h,d,:]
    {
        const int d   = t & 63;
        const int blb = t >> 6;
        float acc[4];
        #pragma unroll
        for (int it = 0; it < 4; ++it) acc[it] = bq[h * DHEAD + d];
        const float* wq = Wq + (h * DHEAD + d) * CIN;
        for (int j = 0; j < CIN; j += 4) {
            float4 w4 = *(const float4*)(wq + j);
            #pragma unroll
            for (int it = 0; it < 4; ++it) {
                const float* qr = &qtile[blb + it * 4][j];
                acc[it] += w4.x * qr[0] + w4.y * qr[1] + w4.z * qr[2] + w4.w * qr[3];
            }
        }
        #pragma unroll
        for (int it = 0; it < 4; ++it) q16[blb + it * 4][d] = acc[it];
    }
    __syncthreads();

    // qb
    if (t < 16) {
        float s = 0.f;
        for (int d = 0; d < DHEAD; ++d) s += q16[t][d] * bk[h * DHEAD + d];
        qb_ws[(b0 + t) * NH + h] = s;
    }

    // qk[bl][i] = sum_d q16[bl][d] * Wk[h,d,i]  (Wk loads fully coalesced)
    {
        const int i4  = (t & 63) * 4;
        const int blb = t >> 6;
        float4 acc[4] = {};
        for (int d = 0; d < DHEAD; ++d) {
            float4 wk4 = *(const float4*)(Wk + (h * DHEAD + d) * CIN + i4);
            #pragma unroll
            for (int it = 0; it < 4; ++it) {
                const float qv = q16[blb + it * 4][d];
                acc[it].x += wk4.x * qv; acc[it].y += wk4.y * qv;
                acc[it].z += wk4.z * qv; acc[it].w += wk4.w * qv;
            }
        }
        #pragma unroll
        for (int it = 0; it < 4; ++it)
            *(float4*)(qk_ws + (((size_t)(b0 + blb + it * 4)) * NH + h) * CIN + i4) = acc[it];
    }
}

// ---------------------------------------------------------------------------
// Kernel 2: one workgroup per batch b (8 wave32).  Streams keys[b] and
// values[b] exactly once (512 MB total over the grid -> ~22us HBM floor).
// Phase 1: scores = keys[b] x qk (f32 WMMA 16x16x4, heads padded to 16)
// Phase 2: softmax per head (wave-per-head, shfl_xor reductions)
// Phase 3: out = attn x values[b] (f32 WMMA 16x16x4)
// ---------------------------------------------------------------------------
#define QK_STRIDE 260   // 4*lane banks mod 64 -> conflict-free b64 reads
#define SC_STRIDE 9     // stride 9 (gcd(9,64)=1) -> conflict-free column walks

__global__ __launch_bounds__(256) void attn_main_kernel(
    const float* __restrict__ keys, const float* __restrict__ values,
    const float* __restrict__ qk_ws, const float* __restrict__ qb_ws,
    float* __restrict__ out)
{
    const int b    = blockIdx.x;
    const int t    = threadIdx.x;
    const int lane = t & 31;
    const int w    = t >> 5;               // wave id 0..7

    __shared__ float qks[16 * QK_STRIDE];  // 16.6 KB (heads 8..15 zero)
    __shared__ float sc[SEQ * SC_STRIDE];  // 36 KB scores / attn weights
    __shared__ float qbs[NH];
    __shared__ float rsum[NH];

    // ---- load qk for this batch into LDS, zero the padded heads ----
    {
        const int hh = t >> 5;
        const int i0 = (t & 31) * 8;
        const float* src = qk_ws + ((size_t)b * NH + hh) * CIN + i0;
        float4 a0 = *(const float4*)(src);
        float4 a1 = *(const float4*)(src + 4);
        *(float4*)(&qks[hh * QK_STRIDE + i0])     = a0;
        *(float4*)(&qks[hh * QK_STRIDE + i0 + 4]) = a1;
        float4 z = {0.f, 0.f, 0.f, 0.f};
        *(float4*)(&qks[(8 + hh) * QK_STRIDE + i0])     = z;
        *(float4*)(&qks[(8 + hh) * QK_STRIDE + i0 + 4]) = z;
    }
    if (t < NH) qbs[t] = qb_ws[(size_t)b * NH + t];
    __syncthreads();

    const int headN = lane & 15;           // N index (phase1) / M index (phase3)
    const int koff  = (lane >> 4) << 1;    // 0 for lanes 0-15, 2 for 16-31
    const float qbv = (headN < NH) ? qbs[headN] : 0.f;

    const float* keyb = keys   + (size_t)b * SEQ * CIN;
    const float* valb = values + (size_t)b * SEQ * CIN;

    // ---------------- Phase 1: scores = keys[b] x qk ----------------
    // A = keys tile (16 rows x 4 K), B = qk (4 x 16 heads), C/D 16x16 f32.
    for (int i = 0; i < 8; ++i) {
        const int tile = w * 8 + i;
        const float* arow = keyb + (size_t)(tile * 16 + headN) * CIN + koff;
        const float* brow = &qks[headN * QK_STRIDE + koff];
        v8f c;
        #pragma unroll
        for (int j = 0; j < 8; ++j) c[j] = qbv;   // fold +q·bk into C
        #pragma unroll 4
        for (int k = 0; k < CIN; k += 4) {
            v2f a  = *(const v2f*)(arow + k);
            v2f bb = *(const v2f*)(brow + k);
            c = __builtin_amdgcn_wmma_f32_16x16x4_f32(
                    false, a, false, bb, (short)0, c, false, false);
        }
        // D: vgpr j -> row (tile*16 + j + 8*half), col = head
        const int rbase = tile * 16 + ((lane >> 4) << 3);
        if (headN < NH) {
            #pragma unroll
            for (int j = 0; j < 8; ++j)
                sc[(rbase + j) * SC_STRIDE + headN] = c[j] * 0.125f;  // 1/sqrt(64)
        }
    }
    __syncthreads();

    // ---------------- Phase 2: softmax (wave w -> head w) ----------------
    {
        float m = -3.4e38f;
        for (int s = lane; s < SEQ; s += 32) m = fmaxf(m, sc[s * SC_STRIDE + w]);
        #pragma unroll
        for (int off = 16; off; off >>= 1) m = fmaxf(m, __shfl_xor(m, off, 32));
        float sum = 0.f;
        for (int s = lane; s < SEQ; s += 32) {
            float e = __expf(sc[s * SC_STRIDE + w] - m);
            sc[s * SC_STRIDE + w] = e;
            sum += e;
        }
        #pragma unroll
        for (int off = 16; off; off >>= 1) sum += __shfl_xor(sum, off, 32);
        if (lane == 0) rsum[w] = 1.0f / sum;
    }
    __syncthreads();

    // ---------------- Phase 3: out = attn x values[b] ----------------
    // A = attn (16-padded heads x 4 s), B = values (4 s x 16 cols).
    const int hclamp = headN & 7;   // rows 8..15 of A duplicate 0..7; D rows 8..15 unused
    for (int nt = 0; nt < 2; ++nt) {
        const int col = (w * 2 + nt) * 16 + (lane & 15);
        v8f c = {};
        #pragma unroll 4
        for (int k = 0; k < SEQ; k += 4) {
            v2f a, bb;
            a.x  = sc[(k + koff)     * SC_STRIDE + hclamp];
            a.y  = sc[(k + koff + 1) * SC_STRIDE + hclamp];
            bb.x = valb[(size_t)(k + koff)     * CIN + col];
            bb.y = valb[(size_t)(k + koff + 1) * CIN + col];
            c = __builtin_amdgcn_wmma_f32_16x16x4_f32(
                    false, a, false, bb, (short)0, c, false, false);
        }
        if (lane < 16) {            // D vgpr j -> head j (valid), col
            #pragma unroll
            for (int j = 0; j < NH; ++j)
                out[((size_t)b * NH + j) * CIN + col] = c[j] * rsum[j];
        }
    }
}

// ---------------------------------------------------------------------------
extern "C" void kernel_launch(void* const* d_in, const int* in_sizes, int n_in,
                              void* d_out, int out_size, void* d_ws, size_t ws_size,
                              hipStream_t stream) {
    const float* queries = (const float*)d_in[0];
    const float* keys    = (const float*)d_in[1];
    const float* values  = (const float*)d_in[2];
    const float* Wq      = (const float*)d_in[3];
    const float* bq      = (const float*)d_in[4];
    const float* Wk      = (const float*)d_in[5];
    const float* bk      = (const float*)d_in[6];
    float* out = (float*)d_out;

    float* qk_ws = (float*)d_ws;                       // B*H*256 floats = 2 MB
    float* qb_ws = qk_ws + (size_t)BATCH * NH * CIN;   // B*H floats = 8 KB

    qk_prep_kernel<<<dim3(NH * (BATCH / 16)), dim3(256), 0, stream>>>(
        queries, Wq, bq, Wk, bk, qk_ws, qb_ws);
    attn_main_kernel<<<dim3(BATCH), dim3(256), 0, stream>>>(
        keys, values, qk_ws, qb_ws, out);
}